// HashGridEncoding_74174085202616
// MI455X (gfx1250) — compile-verified
//
#include <hip/hip_runtime.h>
#include <stdint.h>

typedef __attribute__((ext_vector_type(2))) float v2f;
typedef __attribute__((ext_vector_type(4))) float v4f;
typedef __attribute__((ext_vector_type(4))) int   v4i;

namespace {
constexpr int      NLEV  = 16;
constexpr int      FDIM  = 2;
constexpr unsigned LOG2T = 19;
constexpr unsigned TSIZE = 1u << LOG2T;
constexpr unsigned TMASK = TSIZE - 1u;
constexpr unsigned P1    = 2654435761u;
constexpr unsigned P2    = 805459861u;
constexpr int      BLOCK = 256;
}

#if __has_builtin(__builtin_amdgcn_global_load_async_to_lds_b128)
#define HAVE_ASYNC_LDS 1
#else
#define HAVE_ASYNC_LDS 0
#endif

__global__ __launch_bounds__(BLOCK)
void hashgrid_fwd(const float* __restrict__ xg,
                  const float* __restrict__ emb,
                  float* __restrict__ out,
                  int n)
{
    // Per-level resolutions: 16 * (512/16)^(l/15), truncated (canonical NGP table).
    constexpr int kRes[NLEV] = {16, 20, 25, 32, 40, 50, 64, 80,
                                101, 128, 161, 203, 256, 322, 406, 512};

    __shared__ float sx[BLOCK * 3];

    const int       tid  = threadIdx.x;
    const long long base = (long long)blockIdx.x * BLOCK;
    const long long i    = base + tid;

    // ---- Warm L2 with the embedding table (64 MB << 192 MB L2). One 64-B
    // prefetch per thread covers the whole table across the 1M-thread grid.
    {
        const size_t embBytes = (size_t)NLEV * TSIZE * FDIM * sizeof(float);
        const size_t off      = (size_t)i * 64;
        if (off < embBytes)
            __builtin_prefetch((const char*)emb + off, 0, 1);
    }

    // ---- Stage this block's 256x3 coords into LDS (async path on gfx1250).
    const long long totalf = (long long)n * 3;
    const bool      full   = (base * 3 + (long long)BLOCK * 3) <= totalf; // block-uniform
    if (full) {
#if HAVE_ASYNC_LDS
        if (tid < (BLOCK * 3) / 4) {  // 192 lanes x 16B = 3 KB
            auto gp = (v4i __attribute__((address_space(1))) *)(xg + base * 3 + tid * 4);
            auto lp = (v4i __attribute__((address_space(3))) *)(&sx[tid * 4]);
            __builtin_amdgcn_global_load_async_to_lds_b128(gp, lp, 0, 0);
        }
#if __has_builtin(__builtin_amdgcn_s_wait_asynccnt)
        __builtin_amdgcn_s_wait_asynccnt(0);
#else
        asm volatile("s_wait_asynccnt 0" ::: "memory");
#endif
#else
        for (int k = tid; k < BLOCK * 3; k += BLOCK)
            sx[k] = xg[base * 3 + k];
#endif
    } else {
        for (int k = tid; k < BLOCK * 3; k += BLOCK) {
            const long long g = base * 3 + k;
            sx[k] = (g < totalf) ? xg[g] : 0.0f;
        }
    }
    __syncthreads();

    if (i >= n) return;

    // Stride-3 LDS reads: 3 is coprime with the 64 banks -> conflict-free.
    const float px = sx[tid * 3 + 0];
    const float py = sx[tid * 3 + 1];
    const float pz = sx[tid * 3 + 2];

    float feat0[NLEV];
    float feat1[NLEV];

#pragma unroll
    for (int l = 0; l < NLEV; ++l) {
        const int   r  = kRes[l];
        const float rf = (float)(r - 1);

        const float sxx = px * rf;
        const float syy = py * rf;
        const float szz = pz * rf;

        int ix = (int)sxx; ix = ix < 0 ? 0 : ix; ix = ix > r - 2 ? r - 2 : ix;
        int iy = (int)syy; iy = iy < 0 ? 0 : iy; iy = iy > r - 2 ? r - 2 : iy;
        int iz = (int)szz; iz = iz < 0 ? 0 : iz; iz = iz > r - 2 ? r - 2 : iz;

        const float fx = sxx - (float)ix;
        const float fy = syy - (float)iy;
        const float fz = szz - (float)iz;

        // Incremental XOR-hash terms: corner hash = hx ^ hy ^ hz (2 VALU/corner).
        const unsigned hx0 = (unsigned)ix,       hx1 = hx0 + 1u;
        const unsigned hy0 = (unsigned)iy * P1,  hy1 = hy0 + P1;
        const unsigned hz0 = (unsigned)iz * P2,  hz1 = hz0 + P2;

        const float wx1 = fx, wx0 = 1.0f - fx;
        const float wy1 = fy, wy0 = 1.0f - fy;
        const float wz1 = fz, wz0 = 1.0f - fz;

        const v2f* eb = (const v2f*)(emb + ((size_t)l << (LOG2T + 1)));

        float a0 = 0.0f, a1 = 0.0f;
#pragma unroll
        for (int c = 0; c < 8; ++c) {
            const unsigned h = ((c & 1) ? hx1 : hx0) ^
                               ((c & 2) ? hy1 : hy0) ^
                               ((c & 4) ? hz1 : hz0);
            const v2f   e = eb[h & TMASK];                 // 8-byte b64 gather (L2 hit)
            const float w = ((c & 1) ? wx1 : wx0) *
                            ((c & 2) ? wy1 : wy0) *
                            ((c & 4) ? wz1 : wz0);
            a0 = fmaf(w, e.x, a0);
            a1 = fmaf(w, e.y, a1);
        }
        feat0[l] = a0;
        feat1[l] = a1;
    }

    // ---- 32 contiguous floats per point; stream past L2 with NT b128 stores
    // so the 128 MB output does not evict the L2-resident embedding table.
    v4f* op = (v4f*)(out + (size_t)i * (NLEV * FDIM));
#pragma unroll
    for (int q = 0; q < 8; ++q) {
        v4f v = {feat0[2 * q], feat1[2 * q], feat0[2 * q + 1], feat1[2 * q + 1]};
        __builtin_nontemporal_store(v, op + q);
    }
}

extern "C" void kernel_launch(void* const* d_in, const int* in_sizes, int n_in,
                              void* d_out, int out_size, void* d_ws, size_t ws_size,
                              hipStream_t stream) {
    const float* x   = (const float*)d_in[0];
    const float* emb = (const float*)d_in[1];
    float*       out = (float*)d_out;

    const int n = in_sizes[0] / 3;   // x is [N, 3]
    dim3 grid((n + BLOCK - 1) / BLOCK);
    dim3 block(BLOCK);
    hipLaunchKernelGGL(hashgrid_fwd, grid, block, 0, stream, x, emb, out, n);
}